// CMAPEncoder1_49435073577270
// MI455X (gfx1250) — compile-verified
//
#include <hip/hip_runtime.h>
#include <hip/hip_bf16.h>

typedef __attribute__((ext_vector_type(2))) float v2f;
typedef __attribute__((ext_vector_type(8))) float v8f;

#define IN_F  128
#define OUT_F 64
#define KPAIRS (IN_F / 2)            // 64
#define WELEMS (KPAIRS * OUT_F)      // 4096 float2 per weight matrix

// ---------------- Phase 1: degree / normalization ----------------

__global__ void deg_init_kernel(float* __restrict__ deg, int n) {
    int i = blockIdx.x * blockDim.x + threadIdx.x;
    if (i < n) deg[i] = 1.0f;   // self-loop contributes 1
}

__global__ void deg_edges_kernel(const int* __restrict__ ei, float* __restrict__ deg, int E) {
    int e = blockIdx.x * blockDim.x + threadIdx.x;
    if (e < E) atomicAdd(&deg[ei[(size_t)E + e]], 1.0f);   // dst = col
}

__global__ void deg_rsqrt_kernel(float* __restrict__ deg, int n) {
    int i = blockIdx.x * blockDim.x + threadIdx.x;
    if (i < n) deg[i] = rsqrtf(deg[i]);   // deg >= 1 always (self-loop)
}

// agg[n][:] = x[n][:] * dinv[n]^2   (self-loop term), float4 vectorized
__global__ void agg_init_kernel(const float* __restrict__ x, const float* __restrict__ dinv,
                                float* __restrict__ agg, int n) {
    size_t t = (size_t)blockIdx.x * blockDim.x + threadIdx.x;
    if (t >= (size_t)n * (IN_F / 4)) return;
    int node = (int)(t >> 5);           // IN_F/4 = 32 chunks per node
    int c    = (int)(t & 31);
    float d = dinv[node];
    float coef = d * d;
    float4 v = ((const float4*)(x + (size_t)node * IN_F))[c];
    v.x *= coef; v.y *= coef; v.z *= coef; v.w *= coef;
    ((float4*)(agg + (size_t)node * IN_F))[c] = v;
}

// One wave per edge: agg[col][:] += coef * x[row][:]  (atomics resolve in L2)
__global__ void scatter_kernel(const float* __restrict__ x, const int* __restrict__ ei,
                               const float* __restrict__ dinv, float* __restrict__ agg, int E) {
    size_t t = (size_t)blockIdx.x * blockDim.x + threadIdx.x;
    int e    = (int)(t >> 5);
    int lane = (int)(t & 31);
    if (e >= E) return;
    int r = ei[e];
    int c = ei[(size_t)E + e];
    float coef = dinv[r] * dinv[c];
    const float* xs = x   + (size_t)r * IN_F;
    float*       ag = agg + (size_t)c * IN_F;
#pragma unroll
    for (int j = 0; j < 4; ++j) {
        int f = lane + j * 32;
        atomicAdd(&ag[f], xs[f] * coef);
    }
}

// ---------------- Phase 2: dense GEMM via V_WMMA_F32_16X16X4_F32 ----------------
// Each wave: one 16-row tile, all 64 output cols, both weight matrices (shared A frag).
// A frag (16x4 f32): lane<16 -> row=lane, K={kb,kb+1}; lane>=16 -> row=lane-16, K={kb+2,kb+3}
// B frag (4x16 f32): lane<16 -> col=lane, K={kb,kb+1}; lane>=16 -> col=lane-16, K={kb+2,kb+3}
// Weights staged in LDS K-PAIRED: sW2[kpair*64 + n] = {W[2k][n], W[2k+1][n]}, so each
// B fragment is one contiguous ds_load_b64 (no register re-packing).
// C/D (16x16 f32): VGPR i -> row i (lanes 0-15) / row i+8 (lanes 16-31), col = lane&15

__global__ __launch_bounds__(256) void gcn_gemm_wmma_kernel(
    const float* __restrict__ agg,
    const float* __restrict__ Wmu, const float* __restrict__ bmu,
    const float* __restrict__ Wls, const float* __restrict__ bls,
    float* __restrict__ out_mu, float* __restrict__ out_ls, int nTiles)
{
    __shared__ v2f sW2[2 * WELEMS];   // 64 KB: both weight matrices, K-paired layout
    {
        const int tid = threadIdx.x;
#pragma unroll
        for (int i = 0; i < WELEMS / 256; ++i) {   // 16 iterations
            int idx = i * 256 + tid;               // 0..4095
            int kp  = idx >> 6;                    // kpair
            int n   = idx & 63;
            v2f t;
            t.x = Wmu[(2 * kp) * OUT_F + n];
            t.y = Wmu[(2 * kp + 1) * OUT_F + n];
            sW2[idx] = t;
            v2f u;
            u.x = Wls[(2 * kp) * OUT_F + n];
            u.y = Wls[(2 * kp + 1) * OUT_F + n];
            sW2[WELEMS + idx] = u;
        }
    }
    __syncthreads();

    const int tid  = threadIdx.x;
    const int wave = tid >> 5;
    const int lane = tid & 31;
    const int half = lane >> 4;     // 0: K pair {0,1}, 1: K pair {2,3}
    const int lrow = lane & 15;
    const int tile = blockIdx.x * 8 + wave;
    if (tile >= nTiles) return;
    const int rowBase = tile * 16;

    v8f accMu[4] = {{0.f,0.f,0.f,0.f,0.f,0.f,0.f,0.f}, {0.f,0.f,0.f,0.f,0.f,0.f,0.f,0.f},
                    {0.f,0.f,0.f,0.f,0.f,0.f,0.f,0.f}, {0.f,0.f,0.f,0.f,0.f,0.f,0.f,0.f}};
    v8f accLs[4] = {{0.f,0.f,0.f,0.f,0.f,0.f,0.f,0.f}, {0.f,0.f,0.f,0.f,0.f,0.f,0.f,0.f},
                    {0.f,0.f,0.f,0.f,0.f,0.f,0.f,0.f}, {0.f,0.f,0.f,0.f,0.f,0.f,0.f,0.f}};

    const float* aBase = agg + (size_t)(rowBase + lrow) * IN_F + half * 2;

    for (int kb = 0; kb < IN_F; kb += 4) {
        v2f a;
        a.x = aBase[kb];
        a.y = aBase[kb + 1];
        const v2f* bmuP = sW2 + ((kb >> 1) + half) * OUT_F + lrow;   // kpair row
        const v2f* blsP = bmuP + WELEMS;
#pragma unroll
        for (int ct = 0; ct < 4; ++ct) {
            v2f b = bmuP[ct * 16];      // single ds_load_b64
            accMu[ct] = __builtin_amdgcn_wmma_f32_16x16x4_f32(
                false, a, false, b, (short)0, accMu[ct], false, false);
            v2f b2 = blsP[ct * 16];     // single ds_load_b64
            accLs[ct] = __builtin_amdgcn_wmma_f32_16x16x4_f32(
                false, a, false, b2, (short)0, accLs[ct], false, false);
        }
    }

#pragma unroll
    for (int ct = 0; ct < 4; ++ct) {
        const int n = ct * 16 + lrow;
        const float biasMu = bmu[n];
        const float biasLs = bls[n];
#pragma unroll
        for (int i = 0; i < 8; ++i) {
            const int r = rowBase + i + half * 8;
            out_mu[(size_t)r * OUT_F + n] = accMu[ct][i] + biasMu;
            out_ls[(size_t)r * OUT_F + n] = accLs[ct][i] + biasLs;
        }
    }
}

// Scalar fallback for a non-multiple-of-16 row tail (N=100000 -> unused, but safe).
__global__ void gemm_tail_kernel(const float* __restrict__ agg,
                                 const float* __restrict__ Wmu, const float* __restrict__ bmu,
                                 const float* __restrict__ Wls, const float* __restrict__ bls,
                                 float* __restrict__ out_mu, float* __restrict__ out_ls,
                                 int rowStart, int nRows)
{
    int t = blockIdx.x * blockDim.x + threadIdx.x;
    if (t >= nRows * OUT_F) return;
    int r = rowStart + t / OUT_F;
    int n = t % OUT_F;
    const float* a = agg + (size_t)r * IN_F;
    float su = 0.f, sl = 0.f;
    for (int k = 0; k < IN_F; ++k) {
        float av = a[k];
        su += av * Wmu[k * OUT_F + n];
        sl += av * Wls[k * OUT_F + n];
    }
    out_mu[(size_t)r * OUT_F + n] = su + bmu[n];
    out_ls[(size_t)r * OUT_F + n] = sl + bls[n];
}

extern "C" void kernel_launch(void* const* d_in, const int* in_sizes, int n_in,
                              void* d_out, int out_size, void* d_ws, size_t ws_size,
                              hipStream_t stream) {
    const float* x   = (const float*)d_in[0];
    const int*   ei  = (const int*)d_in[1];
    const float* Wmu = (const float*)d_in[2];
    const float* bmu = (const float*)d_in[3];
    const float* Wls = (const float*)d_in[4];
    const float* bls = (const float*)d_in[5];

    const int N = in_sizes[0] / IN_F;      // 100000
    const int E = in_sizes[1] / 2;         // 3200000

    float* out_mu = (float*)d_out;
    float* out_ls = out_mu + (size_t)N * OUT_F;

    // workspace: agg [N,128] fp32, then deg/dinv [N] fp32  (~51.6 MB)
    float* agg = (float*)d_ws;
    float* deg = agg + (size_t)N * IN_F;

    dim3 blk(256);

    deg_init_kernel<<<(N + 255) / 256, blk, 0, stream>>>(deg, N);
    deg_edges_kernel<<<(E + 255) / 256, blk, 0, stream>>>(ei, deg, E);
    deg_rsqrt_kernel<<<(N + 255) / 256, blk, 0, stream>>>(deg, N);

    size_t initThreads = (size_t)N * (IN_F / 4);
    agg_init_kernel<<<(unsigned)((initThreads + 255) / 256), blk, 0, stream>>>(x, deg, agg, N);

    size_t scatThreads = (size_t)E * 32;
    scatter_kernel<<<(unsigned)((scatThreads + 255) / 256), blk, 0, stream>>>(x, ei, deg, agg, E);

    int nTiles = N / 16;                   // 6250, exact for N=100000
    if (nTiles > 0)
        gcn_gemm_wmma_kernel<<<(nTiles + 7) / 8, blk, 0, stream>>>(
            agg, Wmu, bmu, Wls, bls, out_mu, out_ls, nTiles);

    int rem = N % 16;
    if (rem > 0)
        gemm_tail_kernel<<<(rem * OUT_F + 255) / 256, blk, 0, stream>>>(
            agg, Wmu, bmu, Wls, bls, out_mu, out_ls, nTiles * 16, rem);
}